// GaussianLayer_69441031241789
// MI455X (gfx1250) — compile-verified
//
#include <hip/hip_runtime.h>
#include <hip/hip_bf16.h>

// 21x21 Gaussian depthwise conv, reflect padding, 32 x 1024 x 1024 fp32.
// Kernel is exactly rank-1 => separable: g[j] = K[10][j] / sqrt(K[10][10]).
// Each 1D pass is a banded GEMM (K=36) on V_WMMA_F32_16X16X4_F32 (wave32).
// Tile staging uses GLOBAL_LOAD_ASYNC_TO_LDS_B32 (ASYNCcnt) when available.

typedef float v2f __attribute__((ext_vector_type(2)));
typedef float v8f __attribute__((ext_vector_type(8)));

typedef __attribute__((address_space(1))) int as1_int;  // global
typedef __attribute__((address_space(3))) int as3_int;  // LDS

#define IMG 1024
#define TSTR 69   // LDS stride: {69*i mod 64} conflict-free for 16-row fetches
#define HSTR 44   // LDS stride: conflict-free for pass-2 reads and pass-1 writes

#if defined(__gfx1250__) && __has_builtin(__builtin_amdgcn_global_load_async_to_lds_b32)
#define USE_ASYNC_LDS 1
#endif

__device__ __forceinline__ int reflect_idx(int i) {
  i = (i < 0) ? -i : i;                       // jnp 'reflect': -1 -> 1
  return (i >= IMG) ? (2 * IMG - 2 - i) : i;  // 1024 -> 1022
}

__device__ __forceinline__ v8f wmma_k4(v2f a, v2f b, v8f c) {
  return __builtin_amdgcn_wmma_f32_16x16x4_f32(false, a, false, b, (short)0, c,
                                               false, false);
}

__global__ __launch_bounds__(128, 1) void gauss21_sep_wmma(
    const float* __restrict__ x, const float* __restrict__ kw,
    float* __restrict__ out) {
  __shared__ float Tt[64 * TSTR];  // input tile, rows/cols 0..51 valid
  __shared__ float Hh[64 * HSTR];  // horizontally filtered tile
  __shared__ float gsp[64];        // zero-padded 1D gaussian: gsp[16+j] = g[j]

  const int tid  = threadIdx.x;
  const int lane = tid & 31;
  const int wv   = tid >> 5;   // wave 0..3
  const int half = lane >> 4;  // 0: lanes 0-15, 1: lanes 16-31
  const int lm   = lane & 15;

  int b = blockIdx.x;
  const int tx    = b & 31;
  const int ty    = (b >> 5) & 31;
  const int batch = b >> 10;

  // ---- rank-1 factorization of the 2D kernel, zero-padded for banding ----
  if (tid < 64) {
    float v = 0.0f;
    if (tid >= 16 && tid <= 36) {
      float s = sqrtf(kw[10 * 21 + 10]);
      v = kw[10 * 21 + (tid - 16)] / s;
    }
    gsp[tid] = v;
  }

  // ---- stage 52x52 reflect-padded tile into LDS (async when available) ----
  const int row0 = ty * 32 - 10;
  const int col0 = tx * 32 - 10;
  const float* xb = x + (size_t)batch * IMG * IMG;
  for (int idx = tid; idx < 52 * 52; idx += 128) {
    int r = idx / 52, c = idx - r * 52;
    size_t g = (size_t)reflect_idx(row0 + r) * IMG + reflect_idx(col0 + c);
#ifdef USE_ASYNC_LDS
    __builtin_amdgcn_global_load_async_to_lds_b32(
        (as1_int*)(xb + g), (as3_int*)&Tt[r * TSTR + c],
        /*offset=*/0, /*cpol=*/0);
#else
    Tt[r * TSTR + c] = xb[g];
#endif
  }
#ifdef USE_ASYNC_LDS
#if __has_builtin(__builtin_amdgcn_s_wait_asynccnt)
  __builtin_amdgcn_s_wait_asynccnt(0);
#else
  asm volatile("s_wait_asynccnt 0" ::: "memory");
#endif
#endif
  __syncthreads();

  // ---- pass 1 (horizontal): H[r][n] = sum_k T[r][n0+k]*g[k-n], K=36 ----
  // Wave wv owns row-group r0 = wv*16; two column chunks as two independent
  // accumulator chains (hides WMMA D->C latency, no divergent loop).
  {
    const int r0 = wv * 16;
    const float* trow = &Tt[(r0 + lm) * TSTR];
    v8f acc0 = {}, acc1 = {};
#pragma unroll
    for (int kc = 0; kc < 9; ++kc) {
      const int k0 = kc * 4 + half * 2;  // this lane-half's K pair
      v2f a0, a1, bb;
      // A (16x4 fp32): lanes 0-15 hold K=0,1; lanes 16-31 hold K=2,3
      a0.x = trow[k0];
      a0.y = trow[k0 + 1];
      a1.x = trow[16 + k0];
      a1.y = trow[16 + k0 + 1];
      // B (4x16 fp32): band matrix G[k][n] = g[k-n] (shared by both chains)
      bb.x = gsp[k0 - lm + 16];
      bb.y = gsp[k0 + 1 - lm + 16];
      acc0 = wmma_k4(a0, bb, acc0);
      acc1 = wmma_k4(a1, bb, acc1);
    }
    // D layout: VGPR v -> (M = v + half*8, N = lm)
#pragma unroll
    for (int v = 0; v < 8; ++v) {
      Hh[(r0 + half * 8 + v) * HSTR + lm]        = acc0[v];
      Hh[(r0 + half * 8 + v) * HSTR + (16 + lm)] = acc1[v];
    }
  }
  __syncthreads();

  // ---- pass 2 (vertical): Out[m][n] = sum_k g[k-m]*H[m0+k][n], K=36 ----
  {
    const int m0 = (wv & 1) * 16;
    const int n0 = (wv >> 1) * 16;
    v8f acc = {};
#pragma unroll
    for (int kc = 0; kc < 9; ++kc) {
      const int k0 = kc * 4 + half * 2;
      v2f a, bb;
      a.x = gsp[k0 - lm + 16];  // A[m][k] = g[k-m] (transposed band)
      a.y = gsp[k0 + 1 - lm + 16];
      bb.x = Hh[(m0 + k0) * HSTR + (n0 + lm)];
      bb.y = Hh[(m0 + k0 + 1) * HSTR + (n0 + lm)];
      acc = wmma_k4(a, bb, acc);
    }
    float* ob = out + ((size_t)batch * IMG + (size_t)ty * 32) * IMG + tx * 32;
#pragma unroll
    for (int v = 0; v < 8; ++v)
      ob[(size_t)(m0 + half * 8 + v) * IMG + (n0 + lm)] = acc[v];
  }
}

extern "C" void kernel_launch(void* const* d_in, const int* in_sizes, int n_in,
                              void* d_out, int out_size, void* d_ws, size_t ws_size,
                              hipStream_t stream) {
  const float* x  = (const float*)d_in[0];  // 32*1*1024*1024 fp32
  const float* kw = (const float*)d_in[1];  // 1*1*21*21 fp32
  float* out = (float*)d_out;               // 32*1*1024*1024 fp32
  (void)in_sizes; (void)n_in; (void)out_size; (void)d_ws; (void)ws_size;

  dim3 grid(32 * 32 * 32);  // 32 batches x 32x32 tiles of 32x32 outputs
  gauss21_sep_wmma<<<grid, 128, 0, stream>>>(x, kw, out);
}